// GCN_11501922419433
// MI455X (gfx1250) — compile-verified
//
#include <hip/hip_runtime.h>
#include <hip/hip_bf16.h>

typedef __attribute__((ext_vector_type(2))) float v2f;
typedef __attribute__((ext_vector_type(8))) float v8f;

#define D 128
#define NUM_GRAPHS 64
#define A_STRIDE 132   // 128 + 4 pad: lane m reads As[m*132+k]; 132m mod 64 = 4m -> conflict-free

// ---------------- degree / normalization ----------------
__global__ void deg_init_kernel(float* deg, int N) {
    int i = blockIdx.x * blockDim.x + threadIdx.x;
    if (i < N) deg[i] = 1.0f;  // self-loop
}

__global__ void deg_count_kernel(const int* __restrict__ col, float* deg, int E) {
    int e = blockIdx.x * blockDim.x + threadIdx.x;
    if (e < E) atomicAdd(&deg[col[e]], 1.0f);
}

__global__ void rsqrt_inplace_kernel(float* deg, int N) {
    int i = blockIdx.x * blockDim.x + threadIdx.x;
    if (i < N) deg[i] = rsqrtf(deg[i]);
}

// ---------------- fp32 WMMA GEMM: Y[N,128] = X[N,128] @ W[128,128] ----------------
// Block = 256 threads = 8 waves; covers 16 rows x 128 cols. Wave w owns columns [16w,16w+16).
// A tile (16x128) is staged once in LDS (padded stride) and shared by all 8 waves.
// V_WMMA_F32_16X16X4_F32 fragment layout (ISA 7.12.2):
//   A 16x4 : lane L holds A[M=L&15][K=2*(L>>4)+v], v in {0,1}
//   B 4x16 : lane L holds B[K=2*(L>>4)+v][N=L&15]
//   C/D    : vgpr t, lane L -> C[M=t+8*(L>>4)][N=L&15]
__global__ __launch_bounds__(256) void gemm128_wmma(const float* __restrict__ X,
                                                    const float* __restrict__ W,
                                                    float* __restrict__ Y, int N) {
    __shared__ float As[16 * A_STRIDE];

    const int wave = threadIdx.x >> 5;
    const int lane = threadIdx.x & 31;
    const int m    = lane & 15;
    const int hl   = lane >> 4;          // half-wave select
    const long long row0 = (long long)blockIdx.x * 16;
    const int n0 = wave * 16;

    // Cooperative A-tile load: thread t brings floats [8t, 8t+8) of the 16x128 tile.
    {
        const int t   = threadIdx.x;
        const int rr  = t >> 4;          // row in tile 0..15
        const int cc  = (t & 15) * 8;    // col in tile 0..120
        long long r = row0 + rr;
        if (r >= N) r = N - 1;           // clamp; out-of-range rows never stored
        const float4* src = (const float4*)(X + r * D + cc);
        float4 v0 = src[0];
        float4 v1 = src[1];
        float* dst = &As[rr * A_STRIDE + cc];
        dst[0] = v0.x; dst[1] = v0.y; dst[2] = v0.z; dst[3] = v0.w;
        dst[4] = v1.x; dst[5] = v1.y; dst[6] = v1.z; dst[7] = v1.w;
    }
    __syncthreads();

    const float* arow = &As[m * A_STRIDE];
    v8f c = {0.f, 0.f, 0.f, 0.f, 0.f, 0.f, 0.f, 0.f};

    #pragma unroll 8
    for (int s = 0; s < 32; ++s) {
        const int k = 4 * s + 2 * hl;
        v2f a, b;
        a.x = arow[k];
        a.y = arow[k + 1];
        b.x = W[k * D + n0 + m];
        b.y = W[(k + 1) * D + n0 + m];
        c = __builtin_amdgcn_wmma_f32_16x16x4_f32(
                /*neg_a=*/false, a, /*neg_b=*/false, b,
                /*c_mod=*/(short)0, c, /*reuse_a=*/false, /*reuse_b=*/false);
    }

    if (row0 + 16 <= (long long)N) {
        // uniform fast path: no guards
        float* yb = Y + (row0 + 8 * hl) * D + n0 + m;
        #pragma unroll
        for (int t = 0; t < 8; ++t) yb[(long long)t * D] = c[t];
    } else {
        #pragma unroll
        for (int t = 0; t < 8; ++t) {
            long long orow = row0 + 8 * hl + t;
            if (orow < N) Y[orow * D + n0 + m] = c[t];
        }
    }
}

// ---------------- self-loop init: agg = dis^2 * h ----------------
__global__ void init_agg_kernel(float* __restrict__ agg, const float* __restrict__ h,
                                const float* __restrict__ dis, long long total) {
    long long idx = (long long)blockIdx.x * blockDim.x + threadIdx.x;
    if (idx < total) {
        long long i = idx >> 7;
        float d = dis[i];
        agg[idx] = d * d * h[idx];
    }
}

// ---------------- wave-per-edge scatter: agg[col] += dis[row]*dis[col]*h[row] ----------------
__global__ __launch_bounds__(256) void scatter_edges_kernel(const int* __restrict__ row,
                                                            const int* __restrict__ col,
                                                            const float* __restrict__ dis,
                                                            const float* __restrict__ h,
                                                            float* __restrict__ agg, int E) {
    int e    = blockIdx.x * (blockDim.x >> 5) + (threadIdx.x >> 5);
    int lane = threadIdx.x & 31;
    if (e >= E) return;
    int r = row[e];
    int c = col[e];
    float wgt = dis[r] * dis[c];
    float4 v = ((const float4*)(h + (long long)r * D))[lane];   // 32 lanes x 4 = 128
    float* o = agg + (long long)c * D + lane * 4;
    atomicAdd(o + 0, wgt * v.x);
    atomicAdd(o + 1, wgt * v.y);
    atomicAdd(o + 2, wgt * v.z);
    atomicAdd(o + 3, wgt * v.w);
}

// ---------------- epilogue: out = leaky_relu(agg + bias) ----------------
__global__ void bias_lrelu_kernel(const float* __restrict__ agg, const float* __restrict__ bias,
                                  float* __restrict__ out, long long total) {
    long long idx = (long long)blockIdx.x * blockDim.x + threadIdx.x;
    if (idx < total) {
        float v = agg[idx] + bias[idx & (D - 1)];
        out[idx] = v >= 0.f ? v : 0.01f * v;
    }
}

// ---------------- pooling ----------------
__global__ void zero_kernel(float* p, int n) {
    int i = blockIdx.x * blockDim.x + threadIdx.x;
    if (i < n) p[i] = 0.f;
}

__global__ __launch_bounds__(256) void pool_accum_kernel(const float* __restrict__ h,
                                                         const int* __restrict__ batch,
                                                         float* __restrict__ psum,
                                                         float* __restrict__ pcnt, int N) {
    int i    = blockIdx.x * (blockDim.x >> 5) + (threadIdx.x >> 5);
    int lane = threadIdx.x & 31;
    if (i >= N) return;
    int g = batch[i];
    float4 v = ((const float4*)(h + (long long)i * D))[lane];
    float* o = psum + (long long)g * D + lane * 4;
    atomicAdd(o + 0, v.x);
    atomicAdd(o + 1, v.y);
    atomicAdd(o + 2, v.z);
    atomicAdd(o + 3, v.w);
    if (lane == 0) atomicAdd(&pcnt[g], 1.0f);
}

__global__ __launch_bounds__(128) void pool_final_kernel(const float* __restrict__ psum,
                                                         const float* __restrict__ pcnt,
                                                         const float* __restrict__ fcW,
                                                         const float* __restrict__ fcb,
                                                         float* __restrict__ out) {
    __shared__ float red[D];
    int g = blockIdx.x;
    int t = threadIdx.x;
    float cnt = pcnt[g];
    cnt = cnt > 1.0f ? cnt : 1.0f;
    float val = (psum[g * D + t] / cnt) * fcW[t];
    red[t] = val;
    __syncthreads();
    for (int s = 64; s > 0; s >>= 1) {
        if (t < s) red[t] += red[t + s];
        __syncthreads();
    }
    if (t == 0) out[g] = red[0] + fcb[0];
}

// ---------------- launcher ----------------
extern "C" void kernel_launch(void* const* d_in, const int* in_sizes, int n_in,
                              void* d_out, int out_size, void* d_ws, size_t ws_size,
                              hipStream_t stream) {
    (void)n_in; (void)out_size; (void)ws_size;
    const float* x   = (const float*)d_in[0];
    const int*  edge = (const int*)d_in[1];
    const int*  batc = (const int*)d_in[2];
    const float* W1  = (const float*)d_in[3];
    const float* b1  = (const float*)d_in[4];
    const float* W2  = (const float*)d_in[5];
    const float* b2  = (const float*)d_in[6];
    const float* fcW = (const float*)d_in[7];
    const float* fcb = (const float*)d_in[8];

    const int N = in_sizes[0] / D;
    const int E = in_sizes[1] / 2;
    const int* row = edge;       // sources
    const int* col = edge + E;   // targets

    float* ws   = (float*)d_ws;
    float* dis  = ws;                                 // N   (deg -> dis in place)
    float* bufA = dis + N;                            // N*D
    float* bufB = bufA + (size_t)N * D;               // N*D
    float* psum = bufB + (size_t)N * D;               // 64*128
    float* pcnt = psum + NUM_GRAPHS * D;              // 64

    const long long total = (long long)N * D;
    const int tb = 256;
    const int nBlkN   = (N + tb - 1) / tb;
    const int nBlkE   = (E + tb - 1) / tb;
    const int nBlkEl  = (int)((total + tb - 1) / tb);
    const int nBlkG   = (N + 15) / 16;
    const int nBlkEw  = (E + 7) / 8;   // 8 waves/block, wave per edge
    const int nBlkNw  = (N + 7) / 8;

    // normalization
    deg_init_kernel<<<nBlkN, tb, 0, stream>>>(dis, N);
    deg_count_kernel<<<nBlkE, tb, 0, stream>>>(col, dis, E);
    rsqrt_inplace_kernel<<<nBlkN, tb, 0, stream>>>(dis, N);

    // ---- layer 1 ----
    gemm128_wmma<<<nBlkG, tb, 0, stream>>>(x, W1, bufA, N);
    init_agg_kernel<<<nBlkEl, tb, 0, stream>>>(bufB, bufA, dis, total);
    scatter_edges_kernel<<<nBlkEw, tb, 0, stream>>>(row, col, dis, bufA, bufB, E);
    bias_lrelu_kernel<<<nBlkEl, tb, 0, stream>>>(bufB, b1, bufA, total);

    // ---- layer 2 ----
    gemm128_wmma<<<nBlkG, tb, 0, stream>>>(bufA, W2, bufB, N);
    init_agg_kernel<<<nBlkEl, tb, 0, stream>>>(bufA, bufB, dis, total);
    scatter_edges_kernel<<<nBlkEw, tb, 0, stream>>>(row, col, dis, bufB, bufA, E);
    bias_lrelu_kernel<<<nBlkEl, tb, 0, stream>>>(bufA, b2, bufB, total);

    // ---- pooling + fc ----
    zero_kernel<<<(NUM_GRAPHS * D + NUM_GRAPHS + tb - 1) / tb, tb, 0, stream>>>(psum, NUM_GRAPHS * D + NUM_GRAPHS);
    pool_accum_kernel<<<nBlkNw, tb, 0, stream>>>(bufB, batc, psum, pcnt, N);
    pool_final_kernel<<<NUM_GRAPHS, D, 0, stream>>>(psum, pcnt, fcW, fcb, (float*)d_out);
}